// RGCNLayer_77902116815140
// MI455X (gfx1250) — compile-verified
//
#include <hip/hip_runtime.h>
#include <hip/hip_bf16.h>

// R-GCN with basis decomposition for MI455X (gfx1250, wave32).
// Strategy:
//   1) relW[r] = sum_b coeff[r,b] * basis[b]            (small VALU kernel)
//   2) bin edges by relation (16-aligned bins, -1 pad)  (atomics)
//   3) self term: out = X @ Wself + bias                (fp32 WMMA 16x16x4)
//   4) edge term: per 16-edge tile of one relation,
//      Y = gather(X)[16x64] @ relW[r][64x64] via WMMA, then
//      atomic-add Y * inv_deg[recv] into out rows.      (fp32 WMMA 16x16x4)

typedef __attribute__((ext_vector_type(2))) float v2f;
typedef __attribute__((ext_vector_type(8))) float v8f;

#define WMMA_F32X4(a, b, c) \
    __builtin_amdgcn_wmma_f32_16x16x4_f32(false, (a), false, (b), (short)0, (c), false, false)

static constexpr int F = 64;            // IN_F == OUT_F
static constexpr int NREL = 32;
static constexpr int LDSS = 68;         // padded LDS row stride in floats:
                                        // 68*4=272B rows -> float4-aligned stores,
                                        // bank = (4*row + k) mod 64 -> conflict-free frag reads

// ---------------- small utility kernels ----------------

__global__ void k_fill_i32(int* __restrict__ p, int n, int v) {
    int i = blockIdx.x * blockDim.x + threadIdx.x;
    if (i < n) p[i] = v;
}

__global__ void k_count(const int* __restrict__ recv, const int* __restrict__ rel,
                        int* __restrict__ deg, int* __restrict__ hist, int E) {
    int e = blockIdx.x * blockDim.x + threadIdx.x;
    if (e < E) {
        atomicAdd(&deg[recv[e]], 1);
        atomicAdd(&hist[rel[e]], 1);
    }
}

__global__ void k_prefix(const int* __restrict__ hist, int* __restrict__ cursor, int nrel) {
    if (blockIdx.x == 0 && threadIdx.x == 0) {
        int s = 0;
        for (int r = 0; r < nrel; ++r) {
            cursor[r] = s;
            s = (s + hist[r] + 15) & ~15;   // 16-align every bin start
        }
    }
}

__global__ void k_invdeg(const int* __restrict__ deg, float* __restrict__ invd, int n) {
    int i = blockIdx.x * blockDim.x + threadIdx.x;
    if (i < n) invd[i] = 1.0f / fmaxf((float)deg[i], 1.0f);
}

__global__ void k_scatter(const int* __restrict__ rel, int* __restrict__ cursor,
                          int* __restrict__ binned, int E) {
    int e = blockIdx.x * blockDim.x + threadIdx.x;
    if (e < E) {
        int pos = atomicAdd(&cursor[rel[e]], 1);
        binned[pos] = e;
    }
}

__global__ void k_relw(const float* __restrict__ coeff, const float* __restrict__ basis,
                       float* __restrict__ relW, int nbases) {
    int idx = blockIdx.x * blockDim.x + threadIdx.x;   // r*4096 + i*64 + o
    if (idx >= NREL * F * F) return;
    int r  = idx >> 12;
    int io = idx & 4095;
    float acc = 0.0f;
    for (int b = 0; b < nbases; ++b)
        acc += coeff[r * nbases + b] * basis[(long)b * (F * F) + io];
    relW[idx] = acc;
}

// ---------------- WMMA kernels ----------------

// out[n] = X[n] @ Wself + bias, one 16-node tile per (1-wave) workgroup.
__global__ __launch_bounds__(32) void k_self(const float* __restrict__ nf,
                                             const float* __restrict__ sw,
                                             const float* __restrict__ bias,
                                             float* __restrict__ out, int nNodes) {
    __shared__ float Alds[16 * LDSS];
    int lane = threadIdx.x;
    int base = blockIdx.x * 16;

    for (int jj = 0; jj < 8; ++jj) {               // 16 rows x 16 float4 cols
        int idx = jj * 32 + lane;
        int row = idx >> 4, c4 = idx & 15;
        int n = base + row;
        float4 val = make_float4(0.f, 0.f, 0.f, 0.f);
        if (n < nNodes) val = *(const float4*)(nf + (long)n * F + c4 * 4);
        *(float4*)&Alds[row * LDSS + c4 * 4] = val;
    }
    __syncthreads();

    int half = lane >> 4, l15 = lane & 15;
    v8f acc[4] = {};
#pragma unroll
    for (int ks = 0; ks < 16; ++ks) {              // K = 64 in steps of 4
        int kb = ks * 4 + half * 2;
        v2f a = *(const v2f*)&Alds[l15 * LDSS + kb];
        const float* bp = sw + (long)kb * F + l15;
#pragma unroll
        for (int nb = 0; nb < 4; ++nb) {
            v2f b; b.x = bp[nb * 16]; b.y = bp[nb * 16 + F];
            acc[nb] = WMMA_F32X4(a, b, acc[nb]);
        }
    }
#pragma unroll
    for (int nb = 0; nb < 4; ++nb) {
        int col = nb * 16 + l15;
        float bv = bias[col];
#pragma unroll
        for (int v = 0; v < 8; ++v) {
            int n = base + v + half * 8;
            if (n < nNodes) out[(long)n * F + col] = acc[nb][v] + bv;
        }
    }
}

// One 16-edge tile (single relation, guaranteed by 16-aligned bins) per wave.
__global__ __launch_bounds__(32) void k_edge(const float* __restrict__ nf,
                                             const int* __restrict__ senders,
                                             const int* __restrict__ receivers,
                                             const int* __restrict__ rel,
                                             const float* __restrict__ relW,
                                             const float* __restrict__ invd,
                                             const int* __restrict__ binned,
                                             float* __restrict__ out) {
    __shared__ float Alds[16 * LDSS];
    __shared__ int   sid[16];
    __shared__ int   srecv[16];
    __shared__ float sscale[16];

    int lane = threadIdx.x;
    long base = (long)blockIdx.x * 16;

    int id = -1;
    if (lane < 16) {
        id = binned[base + lane];
        int rcv = -1; float sc = 0.f;
        if (id >= 0) { rcv = receivers[id]; sc = invd[rcv]; }
        sid[lane] = id; srecv[lane] = rcv; sscale[lane] = sc;
    }
    unsigned long long m = __ballot(id >= 0);
    if (m == 0ull) return;                         // wave-uniform: pure padding tile
    int frow = __ffsll((long long)m) - 1;          // lane index == row index
    int r = rel[binned[base + frow]];
    __syncthreads();

    for (int jj = 0; jj < 8; ++jj) {               // gather 16 sender rows
        int idx = jj * 32 + lane;
        int row = idx >> 4, c4 = idx & 15;
        int eid = sid[row];
        float4 val = make_float4(0.f, 0.f, 0.f, 0.f);
        if (eid >= 0) {
            long s = senders[eid];
            val = *(const float4*)(nf + s * F + c4 * 4);
        }
        *(float4*)&Alds[row * LDSS + c4 * 4] = val;
    }
    __syncthreads();

    int half = lane >> 4, l15 = lane & 15;
    const float* Wr = relW + (long)r * F * F;
    v8f acc[4] = {};
#pragma unroll
    for (int ks = 0; ks < 16; ++ks) {
        int kb = ks * 4 + half * 2;
        v2f a = *(const v2f*)&Alds[l15 * LDSS + kb];
        const float* bp = Wr + (long)kb * F + l15;
#pragma unroll
        for (int nb = 0; nb < 4; ++nb) {
            v2f b; b.x = bp[nb * 16]; b.y = bp[nb * 16 + F];
            acc[nb] = WMMA_F32X4(a, b, acc[nb]);
        }
    }
#pragma unroll
    for (int nb = 0; nb < 4; ++nb) {
        int col = nb * 16 + l15;
#pragma unroll
        for (int v = 0; v < 8; ++v) {
            int row = v + half * 8;
            int rcv = srecv[row];
            if (rcv >= 0)
                unsafeAtomicAdd(&out[(long)rcv * F + col], acc[nb][v] * sscale[row]);
        }
    }
}

// ---------------- host launch ----------------

extern "C" void kernel_launch(void* const* d_in, const int* in_sizes, int n_in,
                              void* d_out, int out_size, void* d_ws, size_t ws_size,
                              hipStream_t stream) {
    const float* nf    = (const float*)d_in[0];
    const int*   send  = (const int*)d_in[1];
    const int*   recv  = (const int*)d_in[2];
    const int*   rel   = (const int*)d_in[3];
    const float* basis = (const float*)d_in[4];
    const float* coeff = (const float*)d_in[5];
    const float* sw    = (const float*)d_in[6];
    const float* bias  = (const float*)d_in[7];
    float* out = (float*)d_out;

    const int N  = in_sizes[0] / F;
    const int E  = in_sizes[1];
    const int NB = in_sizes[4] / (F * F);
    const int EPAD = ((E + 15) / 16) * 16 + NREL * 16;   // bin alignment slack

    char* ws = (char*)d_ws;
    size_t off = 0;
    auto carve = [&](size_t bytes) -> char* {
        char* p = ws + off;
        off = (off + bytes + 255) & ~(size_t)255;
        return p;
    };
    float* relW   = (float*)carve(sizeof(float) * NREL * F * F);
    int*   deg    = (int*)  carve(sizeof(int) * N);
    float* invd   = (float*)carve(sizeof(float) * N);
    int*   hist   = (int*)  carve(sizeof(int) * NREL);
    int*   cursor = (int*)  carve(sizeof(int) * NREL);
    int*   binned = (int*)  carve(sizeof(int) * EPAD);
    (void)ws_size; (void)n_in; (void)out_size;

    k_fill_i32<<<(N + 255) / 256, 256, 0, stream>>>(deg, N, 0);
    k_fill_i32<<<1, 256, 0, stream>>>(hist, NREL, 0);
    k_fill_i32<<<(EPAD + 255) / 256, 256, 0, stream>>>(binned, EPAD, -1);
    k_count<<<(E + 255) / 256, 256, 0, stream>>>(recv, rel, deg, hist, E);
    k_prefix<<<1, 1, 0, stream>>>(hist, cursor, NREL);
    k_invdeg<<<(N + 255) / 256, 256, 0, stream>>>(deg, invd, N);
    k_scatter<<<(E + 255) / 256, 256, 0, stream>>>(rel, cursor, binned, E);
    k_relw<<<(NREL * F * F + 255) / 256, 256, 0, stream>>>(coeff, basis, relW, NB);
    k_self<<<(N + 15) / 16, 32, 0, stream>>>(nf, sw, bias, out, N);
    k_edge<<<EPAD / 16, 32, 0, stream>>>(nf, send, recv, rel, relW, invd, binned, out);
}